// GraphSAGEModel_72619307041226
// MI455X (gfx1250) — compile-verified
//
#include <hip/hip_runtime.h>
#include <hip/hip_bf16.h>

#define N_NODES 100000
#define N_EDGES 1600000
#define D 64            // D_IN == D_OUT == 64

typedef float v2f __attribute__((ext_vector_type(2)));
typedef float v8f __attribute__((ext_vector_type(8)));

// ---------------------------------------------------------------------------
// Kernel 1: acc[i][:] = x[i][:]  (self-loop term), deg[i] = 1.0 (self-loop +1)
// Re-initialized on every launch -> deterministic across graph replays.
// ---------------------------------------------------------------------------
__global__ void sage_init(const float* __restrict__ x,
                          float* __restrict__ acc,
                          float* __restrict__ deg) {
    const int gid = blockIdx.x * blockDim.x + threadIdx.x;   // N_NODES*16 threads
    if (gid < N_NODES * (D / 4)) {
        ((float4*)acc)[gid] = ((const float4*)x)[gid];
    }
    if (gid < N_NODES) {
        deg[gid] = 1.0f;
    }
}

// ---------------------------------------------------------------------------
// Kernel 2: for each edge e: acc[dst[e]][:] += x[src[e]][:]; deg[dst[e]] += 1
// 16 threads per edge, each: one 128-bit gather load + 4 f32 atomic adds.
// Whole working set is L2-resident (x 25.6MB + acc 25.6MB << 192MB L2), so
// these atomics run at L2 atomic-unit throughput.
// ---------------------------------------------------------------------------
__global__ void sage_scatter(const float* __restrict__ x,
                             const int* __restrict__ src,
                             const int* __restrict__ dst,
                             float* __restrict__ acc,
                             float* __restrict__ deg) {
    const unsigned gid = blockIdx.x * blockDim.x + threadIdx.x;
    const unsigned e = gid >> 4;       // edge index
    const unsigned c = gid & 15;       // 4-float chunk within the 64-f feature
    if (e >= N_EDGES) return;

    const int s = src[e];
    const int d = dst[e];

    const float4 v = *(const float4*)(x + (size_t)s * D + c * 4);
    float* a = acc + (size_t)d * D + c * 4;

    __hip_atomic_fetch_add(a + 0, v.x, __ATOMIC_RELAXED, __HIP_MEMORY_SCOPE_AGENT);
    __hip_atomic_fetch_add(a + 1, v.y, __ATOMIC_RELAXED, __HIP_MEMORY_SCOPE_AGENT);
    __hip_atomic_fetch_add(a + 2, v.z, __ATOMIC_RELAXED, __HIP_MEMORY_SCOPE_AGENT);
    __hip_atomic_fetch_add(a + 3, v.w, __ATOMIC_RELAXED, __HIP_MEMORY_SCOPE_AGENT);

    if (c == 0) {
        __hip_atomic_fetch_add(deg + d, 1.0f, __ATOMIC_RELAXED, __HIP_MEMORY_SCOPE_AGENT);
    }
}

// ---------------------------------------------------------------------------
// Kernel 3: out = (acc @ W^T) / deg + b   using V_WMMA_F32_16X16X4_F32.
// Block = 128 threads = 4 waves. Block owns a 16-row M tile; wave w owns the
// 16-column N tile [16w, 16w+16). K loop: 16 x (16x16x4) fp32 WMMAs.
//
// ISA VGPR layouts (CDNA5 ISA):
//   A (16x4, MxK):  lanes 0-15 -> M=lane,   VGPR j -> K = j        (K half 0)
//                   lanes 16-31-> M=lane-16,VGPR j -> K = 2 + j    (K half 1)
//   B (4x16, KxN):  lanes 0-15 -> N=lane,   VGPR j -> K = j
//                   lanes 16-31-> N=lane-16,VGPR j -> K = 2 + j
//   C/D (16x16):    lanes 0-15 -> N=lane,   VGPR r -> M = r
//                   lanes 16-31-> N=lane-16,VGPR r -> M = 8 + r
// ---------------------------------------------------------------------------
__global__ void sage_gemm(const float* __restrict__ acc,
                          const float* __restrict__ deg,
                          const float* __restrict__ W,   // [D_OUT, D_IN] row-major
                          const float* __restrict__ bias,
                          float* __restrict__ out) {
    const int lane  = threadIdx.x & 31;
    const int wave  = threadIdx.x >> 5;       // 0..3 -> N tile
    const int tileM = blockIdx.x * 16;        // 100000 / 16 = 6250 exactly
    const int tileN = wave * 16;

    const int half  = lane >> 4;              // 0 or 1: which K pair / M half
    const int l15   = lane & 15;
    const int kb    = half * 2;               // K offset inside a K=4 step

    // A source: row of the accumulator this lane feeds (h_neigh numerator)
    const float* arow = acc + (size_t)(tileM + l15) * D;
    // B source: out[m][n] = sum_k acc[m][k] * W[n][k]  ->  B[k][n] = W[n][k]
    const float* wrow = W + (size_t)(tileN + l15) * D;

    v8f c = {};
#pragma unroll
    for (int kk = 0; kk < D; kk += 4) {
        v2f a = *(const v2f*)(arow + kk + kb);
        v2f b = *(const v2f*)(wrow + kk + kb);
        c = __builtin_amdgcn_wmma_f32_16x16x4_f32(
                /*neg_a=*/false, a, /*neg_b=*/false, b,
                /*c_mod=*/(short)0, c, /*reuse_a=*/false, /*reuse_b=*/false);
    }

    // Epilogue: out[m][n] = c / deg[m] + bias[n]
    const float bn   = bias[tileN + l15];
    const int mbase  = tileM + 8 * half;
#pragma unroll
    for (int r = 0; r < 8; ++r) {
        const int m = mbase + r;
        out[(size_t)m * D + tileN + l15] = c[r] / deg[m] + bn;
    }
}

// ---------------------------------------------------------------------------
extern "C" void kernel_launch(void* const* d_in, const int* in_sizes, int n_in,
                              void* d_out, int out_size, void* d_ws, size_t ws_size,
                              hipStream_t stream) {
    const float* x   = (const float*)d_in[0];   // [N_NODES, 64]
    const int*   src = (const int*)  d_in[1];   // [N_EDGES]
    const int*   dst = (const int*)  d_in[2];   // [N_EDGES]
    const float* W   = (const float*)d_in[3];   // [64, 64]
    const float* b   = (const float*)d_in[4];   // [64]
    float*       out = (float*)d_out;           // [N_NODES, 64]

    // workspace: acc [N_NODES*64] f32, deg [N_NODES] f32  (~26 MB)
    float* acc = (float*)d_ws;
    float* deg = acc + (size_t)N_NODES * D;

    {   // init: N_NODES*16 float4 copies + N_NODES deg writes
        const int threads = 256;
        const int total   = N_NODES * (D / 4);
        const int blocks  = (total + threads - 1) / threads;   // 6250
        sage_init<<<blocks, threads, 0, stream>>>(x, acc, deg);
    }
    {   // scatter: 16 threads per edge
        const int threads = 256;
        const long long total = (long long)N_EDGES * 16;       // 25.6M
        const int blocks  = (int)((total + threads - 1) / threads); // 100000
        sage_scatter<<<blocks, threads, 0, stream>>>(x, src, dst, acc, deg);
    }
    {   // gemm: one block per 16-row tile, 4 waves cover the 64 output cols
        const int blocks  = N_NODES / 16;                      // 6250
        sage_gemm<<<blocks, 128, 0, stream>>>(acc, deg, W, b, out);
    }
}